// TransPool_55362128445546
// MI455X (gfx1250) — compile-verified
//
#include <hip/hip_runtime.h>
#include <hip/hip_bf16.h>
#include <stdint.h>

// ---------------------------------------------------------------------------
// TransPool for MI455X (gfx1250, wave32). fp32 everywhere (memory-bound; the
// only GEMM is 50000x128x128 = 1.6 GFLOP -> use V_WMMA_F32_16X16X4_F32).
// ---------------------------------------------------------------------------

#define DEV __device__ __forceinline__

constexpr int N  = 50000;
constexpr int E  = 600000;
constexpr int F  = 128;          // IN == OUT
constexpr int K  = 42500;        // ceil(0.85 * N)
constexpr int NC = E + K;        // candidate assignment edges
constexpr int SORT_N = 65536;    // next pow2 >= N for bitonic sort

// flat output layout (reference tuple order, everything as float)
constexpr long long O0 = 0;                       // x_out        K*F
constexpr long long O1 = O0 + (long long)K * F;   // out_edge_index 2*E
constexpr long long O2 = O1 + 2LL * E;            // out_w        E
constexpr long long O3 = O2 + (long long)E;       // batch[sup]   K
constexpr long long O4 = O3 + (long long)K;       // super_nodes  K
constexpr long long O5 = O4 + (long long)K;       // assign_index 2*NC
constexpr long long O6 = O5 + 2LL * NC;           // aw           NC

typedef float v2f __attribute__((ext_vector_type(2)));
typedef float v8f __attribute__((ext_vector_type(8)));

// ---- helpers ---------------------------------------------------------------

DEV unsigned ord_u32(float f) {              // monotone float -> uint
  unsigned u = __float_as_uint(f);
  return (u & 0x80000000u) ? ~u : (u | 0x80000000u);
}
DEV float inv_ord(unsigned e) {
  return __uint_as_float((e & 0x80000000u) ? (e ^ 0x80000000u) : ~e);
}

DEV float gumbel_of(unsigned i) {            // deterministic per-index gumbel
  unsigned long long zx = (unsigned long long)i * 0x9E3779B97F4A7C15ULL
                          + 0x123456789ABCDEF0ULL;
  zx ^= zx >> 30; zx *= 0xBF58476D1CE4E5B9ULL;
  zx ^= zx >> 27; zx *= 0x94D049BB133111EBULL;
  zx ^= zx >> 31;
  float u = (float)((zx >> 40) + 1ULL) * (1.0f / 16777218.0f); // in (0,1)
  return -logf(-logf(u));
}

DEV void cand_of(int ci, const int* __restrict__ srcv, const int* __restrict__ dstv,
                 const int* __restrict__ node2sup, const int* __restrict__ sup_idx,
                 int& cs, int& ks) {
  if (ci < E) { cs = srcv[ci]; ks = node2sup[dstv[ci]]; }
  else        { cs = sup_idx[ci - E]; ks = ci - E; }
}

// ---- fill kernels ----------------------------------------------------------

__global__ void fill_f32_kernel(float* p, float v, int n) {
  int i = blockIdx.x * blockDim.x + threadIdx.x;
  if (i < n) p[i] = v;
}
__global__ void fill_u32_kernel(unsigned* p, unsigned v, int n) {
  int i = blockIdx.x * blockDim.x + threadIdx.x;
  if (i < n) p[i] = v;
}
__global__ void fill_ord_kernel(unsigned* p, float v, int n) {
  int i = blockIdx.x * blockDim.x + threadIdx.x;
  if (i < n) p[i] = ord_u32(v);
}

// ---- stage 1: score GCN ----------------------------------------------------

// h1[i] = dot(x[i,:], score_W)   (wave-per-node, shfl reduce)
__global__ void score_h1_kernel(const float* __restrict__ x,
                                const float* __restrict__ sW,
                                float* __restrict__ h1) {
  int node = blockIdx.x * 8 + (threadIdx.x >> 5);
  int lane = threadIdx.x & 31;
  if (node >= N) return;
  const float* xr = x + (size_t)node * F + lane * 4;
  float d = xr[0]*sW[lane*4+0] + xr[1]*sW[lane*4+1]
          + xr[2]*sW[lane*4+2] + xr[3]*sW[lane*4+3];
  for (int o = 16; o > 0; o >>= 1) d += __shfl_down(d, o, 32);
  if (lane == 0) h1[node] = d;
}

__global__ void deg_w0_kernel(const int* __restrict__ srcv, const int* __restrict__ dstv,
                              const float* __restrict__ w, float* __restrict__ deg0) {
  int e = blockIdx.x * blockDim.x + threadIdx.x;
  if (e >= E) return;
  int s = srcv[e], d = dstv[e];
  float w0 = (s == d) ? 0.0f : w[e];
  if (w0 != 0.0f) atomicAdd(&deg0[d], w0);
}

__global__ void score_agg_kernel(const int* __restrict__ srcv, const int* __restrict__ dstv,
                                 const float* __restrict__ w, const float* __restrict__ deg0,
                                 const float* __restrict__ h1, float* __restrict__ score) {
  int e = blockIdx.x * blockDim.x + threadIdx.x;
  if (e >= E) return;
  int s = srcv[e], d = dstv[e];
  float w0 = (s == d) ? 0.0f : w[e];
  if (w0 == 0.0f) return;
  float nm = rsqrtf(deg0[s] + 1.0f) * w0 * rsqrtf(deg0[d] + 1.0f);
  atomicAdd(&score[d], nm * h1[s]);
}

// finalize score + build 64-bit sort keys: (ordered(score) << 32) | ~index
__global__ void score_final_kernel(float* __restrict__ score, const float* __restrict__ h1,
                                   const float* __restrict__ deg0, const float* __restrict__ sB,
                                   unsigned long long* __restrict__ keys) {
  int i = blockIdx.x * blockDim.x + threadIdx.x;
  if (i >= SORT_N) return;
  if (i < N) {
    float s = score[i] + h1[i] / (deg0[i] + 1.0f) + sB[0];
    score[i] = s;
    keys[i] = ((unsigned long long)ord_u32(s) << 32) | (unsigned)(~(unsigned)i);
  } else {
    keys[i] = 0ULL;   // pad: sorts last under descending order
  }
}

// ---- stage 2: exact top-K via bitonic sort (descending) --------------------

__global__ void bitonic_step_kernel(unsigned long long* __restrict__ keys, int j, int kk) {
  int i = blockIdx.x * blockDim.x + threadIdx.x;
  int ixj = i ^ j;
  if (ixj > i) {
    unsigned long long a = keys[i], b = keys[ixj];
    bool asc = ((i & kk) == 0);
    if ((a < b) == asc) { keys[i] = b; keys[ixj] = a; }  // overall descending
  }
}

__global__ void extract_topk_kernel(const unsigned long long* __restrict__ keys,
                                    int* __restrict__ sup_idx, int* __restrict__ is_sup,
                                    int* __restrict__ node2sup, const int* __restrict__ batch,
                                    float* __restrict__ out) {
  int r = blockIdx.x * blockDim.x + threadIdx.x;
  if (r >= K) return;
  int i = (int)(~(unsigned)(keys[r] & 0xFFFFFFFFull));
  sup_idx[r]  = i;
  is_sup[i]   = 1;
  node2sup[i] = r;
  out[O4 + r] = (float)i;
  out[O3 + r] = (float)batch[i];
}

// ---- stage 3: h = x @ aggr_W via V_WMMA_F32_16X16X4_F32 --------------------
// Block = 256 thr (8 waves). Each block: one 16-row tile of x staged in LDS
// (stride 132 floats -> conflict-free per-lane row reads); each wave computes
// a 16x16 output tile over K=128 with 32 WMMA f32 16x16x4 steps, FULLY
// unrolled so the compiler can clause all A ds_loads / B global_loads ahead of
// the serial WMMA accumulate chain. B fragments stream from global (aggr_W is
// 64 KB, L2-resident). N % 16 == 0, no guards, EXEC all-ones around WMMA.
__global__ void __launch_bounds__(256) gemm_h_kernel(const float* __restrict__ x,
                                                     const float* __restrict__ Wg,
                                                     float* __restrict__ h) {
  __shared__ float As[16 * 132];
  const int tile = blockIdx.x;
  const int t = threadIdx.x;
#pragma unroll
  for (int it = 0; it < 8; ++it) {           // 16x128 A tile, coalesced
    int idx = t + it * 256;
    int r = idx >> 7, k = idx & 127;
    As[r * 132 + k] = x[((size_t)tile * 16 + r) * F + k];
  }
  __syncthreads();

  const int wave = t >> 5;
  const int lane = t & 31;
  const int hi   = lane >> 4;                // lane half selects K 0/1 vs 2/3
  const int m    = lane & 15;
  const int col  = wave * 16 + m;

  v8f c = {};
#pragma unroll
  for (int kk = 0; kk < 32; ++kk) {
    v2f a, b;
    // A 16x4 f32 layout: lanes 0-15 -> {K0,K1}, lanes 16-31 -> {K2,K3}
    a.x = As[m * 132 + kk * 4 + hi * 2 + 0];
    a.y = As[m * 132 + kk * 4 + hi * 2 + 1];
    // B 4x16 f32 layout: v0 -> rows K0|K1 across lane halves, v1 -> K2|K3
    b.x = Wg[(kk * 4 + hi    ) * F + col];
    b.y = Wg[(kk * 4 + hi + 2) * F + col];
    c = __builtin_amdgcn_wmma_f32_16x16x4_f32(false, a, false, b,
                                              (short)0, c, false, false);
  }
  // C/D layout: VGPR r -> M=r (lanes 0-15), M=r+8 (lanes 16-31)
  const size_t row0 = (size_t)tile * 16 + hi * 8;
#pragma unroll
  for (int r = 0; r < 8; ++r)
    h[(row0 + r) * F + col] = c[r];
}

// ---- stage 4: aggregation GCN (temp_x) -------------------------------------

__global__ void deg_w1_kernel(const int* __restrict__ srcv, const int* __restrict__ dstv,
                              const float* __restrict__ w, const int* __restrict__ is_sup,
                              float* __restrict__ w1buf, float* __restrict__ deg1) {
  int e = blockIdx.x * blockDim.x + threadIdx.x;
  if (e >= E) return;
  int s = srcv[e], d = dstv[e];
  float w1 = (is_sup[d] && s != d) ? w[e] : 0.0f;
  w1buf[e] = w1;
  if (w1 != 0.0f) atomicAdd(&deg1[d], w1);
}

__global__ void agg_edges_kernel(const int* __restrict__ srcv, const int* __restrict__ dstv,
                                 const float* __restrict__ w1buf, const float* __restrict__ deg1,
                                 const float* __restrict__ h, float* __restrict__ agg) {
  int tid = blockIdx.x * blockDim.x + threadIdx.x;
  if (tid >= E * 32) return;
  int e = tid >> 5;
  int c = (tid & 31) * 4;
  float w1 = w1buf[e];
  if (w1 == 0.0f) return;
  int s = srcv[e], d = dstv[e];
  float nm = rsqrtf(deg1[s] + 1.0f) * w1 * rsqrtf(deg1[d] + 1.0f);
  const float* hs = h   + (size_t)s * F + c;
  float*       ad = agg + (size_t)d * F + c;
  atomicAdd(ad + 0, nm * hs[0]);
  atomicAdd(ad + 1, nm * hs[1]);
  atomicAdd(ad + 2, nm * hs[2]);
  atomicAdd(ad + 3, nm * hs[3]);
}

__global__ void tempx_final_kernel(float* __restrict__ tempx, const float* __restrict__ h,
                                   const float* __restrict__ deg1, const float* __restrict__ aB) {
  int i = blockIdx.x * blockDim.x + threadIdx.x;
  if (i >= N * F) return;
  int node = i >> 7, c = i & 127;
  tempx[i] = tempx[i] + h[i] / (deg1[node] + 1.0f) + aB[c];
}

__global__ void supx_kernel(const float* __restrict__ tempx, const int* __restrict__ sup_idx,
                            float* __restrict__ supx) {
  int i = blockIdx.x * blockDim.x + threadIdx.x;
  if (i >= K * F) return;
  int k = i >> 7, c = i & 127;
  float v = tempx[(size_t)sup_idx[k] * F + c];
  supx[i] = v > 0.0f ? v : 0.0f;
}

// ---- stage 5: coarsened graph + assign_index -------------------------------

__global__ void coarse_edges_kernel(const int* __restrict__ srcv, const int* __restrict__ dstv,
                                    const float* __restrict__ w, const int* __restrict__ is_sup,
                                    const int* __restrict__ node2sup, float* __restrict__ out) {
  int e = blockIdx.x * blockDim.x + threadIdx.x;
  if (e >= E) return;
  int s = srcv[e], d = dstv[e];
  int os = node2sup[s], od = node2sup[d];
  bool m1 = is_sup[d] && (s != d);
  bool m2 = m1 && is_sup[s];
  out[O1 + e]     = (float)os;
  out[O1 + E + e] = (float)od;
  out[O2 + e]     = (m2 && os != od) ? w[e] : 0.0f;
  out[O5 + e]      = (float)od;   // cand_sup[e]
  out[O5 + NC + e] = (float)s;    // cand_src[e]
}

__global__ void assign_tail_kernel(const int* __restrict__ sup_idx, float* __restrict__ out) {
  int r = blockIdx.x * blockDim.x + threadIdx.x;
  if (r >= K) return;
  out[O5 + E + r]      = (float)r;
  out[O5 + NC + E + r] = (float)sup_idx[r];
}

// ---- stage 6: assignment weights ------------------------------------------

__global__ void aw_dot_kernel(const float* __restrict__ tempx, const float* __restrict__ supx,
                              const int* __restrict__ srcv, const int* __restrict__ dstv,
                              const int* __restrict__ node2sup, const int* __restrict__ sup_idx,
                              const float* __restrict__ w1buf,
                              float* __restrict__ awbuf, float* __restrict__ rowsum) {
  int ci = blockIdx.x * 8 + (threadIdx.x >> 5);
  int lane = threadIdx.x & 31;
  if (ci >= NC) return;
  int cs, ks; float cw;
  if (ci < E) { cs = srcv[ci]; ks = node2sup[dstv[ci]]; cw = w1buf[ci]; }
  else        { cs = sup_idx[ci - E]; ks = ci - E; cw = 1.0f; }
  const float* a = tempx + (size_t)cs * F + lane * 4;
  const float* b = supx  + (size_t)ks * F + lane * 4;
  float dot = a[0]*b[0] + a[1]*b[1] + a[2]*b[2] + a[3]*b[3];
  for (int o = 16; o > 0; o >>= 1) dot += __shfl_down(dot, o, 32);
  if (lane == 0) {
    float v = cw / (1.0f + expf(-dot));
    awbuf[ci] = v;
    if (v != 0.0f) atomicAdd(&rowsum[cs], v);
  }
}

__global__ void aw_norm_logit_kernel(const int* __restrict__ srcv, const int* __restrict__ dstv,
                                     const int* __restrict__ node2sup, const int* __restrict__ sup_idx,
                                     const float* __restrict__ rowsum, const float* __restrict__ temp,
                                     float* __restrict__ awbuf, float* __restrict__ logitbuf,
                                     unsigned* __restrict__ rowmax) {
  int i = blockIdx.x * blockDim.x + threadIdx.x;
  if (i >= NC) return;
  int cs, ks; cand_of(i, srcv, dstv, node2sup, sup_idx, cs, ks);
  float rs = rowsum[cs];
  float a = awbuf[i] / (rs > 0.0f ? rs : 1.0f);
  awbuf[i] = a;
  float l = -1e30f;
  if (a > 0.0f) l = (logf(fmaxf(a, 1e-30f)) + gumbel_of((unsigned)i)) / temp[0];
  logitbuf[i] = l;
  atomicMax(&rowmax[cs], ord_u32(l));
}

__global__ void aw_exp_kernel(const int* __restrict__ srcv, const int* __restrict__ dstv,
                              const int* __restrict__ node2sup, const int* __restrict__ sup_idx,
                              const unsigned* __restrict__ rowmax,
                              const float* __restrict__ awbuf, float* __restrict__ logitbuf,
                              float* __restrict__ z) {
  int i = blockIdx.x * blockDim.x + threadIdx.x;
  if (i >= NC) return;
  int cs, ks; cand_of(i, srcv, dstv, node2sup, sup_idx, cs, ks);
  float e = 0.0f;
  if (awbuf[i] > 0.0f) e = expf(logitbuf[i] - inv_ord(rowmax[cs]));
  logitbuf[i] = e;
  if (e != 0.0f) atomicAdd(&z[cs], e);
}

__global__ void aw_final_kernel(const int* __restrict__ srcv, const int* __restrict__ dstv,
                                const int* __restrict__ node2sup, const int* __restrict__ sup_idx,
                                const float* __restrict__ z, const float* __restrict__ logitbuf,
                                float* __restrict__ awbuf, float* __restrict__ out) {
  int i = blockIdx.x * blockDim.x + threadIdx.x;
  if (i >= NC) return;
  int cs, ks; cand_of(i, srcv, dstv, node2sup, sup_idx, cs, ks);
  float zz = z[cs];
  float a = logitbuf[i] / (zz > 0.0f ? zz : 1.0f);
  a = (a >= 0.5f / (float)K) ? a : 0.0f;
  awbuf[i] = a;
  out[O6 + i] = a;
}

__global__ void xagg_kernel(const int* __restrict__ srcv, const int* __restrict__ dstv,
                            const int* __restrict__ node2sup, const int* __restrict__ sup_idx,
                            const float* __restrict__ awbuf, const float* __restrict__ tempx,
                            float* __restrict__ xagg) {
  int tid = blockIdx.x * blockDim.x + threadIdx.x;
  if (tid >= NC * 32) return;
  int ci = tid >> 5;
  int c = (tid & 31) * 4;
  float a = awbuf[ci];
  if (a == 0.0f) return;
  int cs, ks; cand_of(ci, srcv, dstv, node2sup, sup_idx, cs, ks);
  const float* tp = tempx + (size_t)cs * F + c;
  float*       xp = xagg  + (size_t)ks * F + c;
  atomicAdd(xp + 0, a * tp[0]);
  atomicAdd(xp + 1, a * tp[1]);
  atomicAdd(xp + 2, a * tp[2]);
  atomicAdd(xp + 3, a * tp[3]);
}

__global__ void xout_kernel(const float* __restrict__ supx, const float* __restrict__ xagg,
                            float* __restrict__ out) {
  int i = blockIdx.x * blockDim.x + threadIdx.x;
  if (i >= K * F) return;
  out[O0 + i] = (supx[i] + xagg[i]) * 0.5f;
}

// ---------------------------------------------------------------------------

#define CDIV(a, b) (((a) + (b) - 1) / (b))

extern "C" void kernel_launch(void* const* d_in, const int* in_sizes, int n_in,
                              void* d_out, int out_size, void* d_ws, size_t ws_size,
                              hipStream_t stream) {
  (void)in_sizes; (void)n_in; (void)out_size; (void)ws_size;

  const float* x      = (const float*)d_in[0];
  const int*   ei     = (const int*)d_in[1];
  const int*   srcv   = ei;
  const int*   dstv   = ei + E;
  const float* w      = (const float*)d_in[2];
  const int*   batch  = (const int*)d_in[3];
  const float* temp   = (const float*)d_in[4];
  const float* scoreW = (const float*)d_in[5];
  const float* scoreB = (const float*)d_in[6];
  const float* aggrW  = (const float*)d_in[7];
  const float* aggrB  = (const float*)d_in[8];
  float* out = (float*)d_out;

  // carve workspace (256B aligned)
  char* base = (char*)d_ws;
  size_t off = 0;
  auto carve = [&](size_t bytes) -> void* {
    void* q = base + off;
    off += (bytes + 255) & ~(size_t)255;
    return q;
  };
  float*    h1       = (float*)carve((size_t)N * 4);
  float*    deg0     = (float*)carve((size_t)N * 4);
  float*    score    = (float*)carve((size_t)N * 4);
  float*    deg1     = (float*)carve((size_t)N * 4);
  float*    rowsum   = (float*)carve((size_t)N * 4);
  unsigned* rowmax   = (unsigned*)carve((size_t)N * 4);
  float*    zbuf     = (float*)carve((size_t)N * 4);
  int*      is_sup   = (int*)carve((size_t)N * 4);
  int*      node2sup = (int*)carve((size_t)N * 4);
  int*      sup_idx  = (int*)carve((size_t)K * 4);
  unsigned long long* keys = (unsigned long long*)carve((size_t)SORT_N * 8);
  float*    w1buf    = (float*)carve((size_t)E * 4);
  float*    awbuf    = (float*)carve((size_t)NC * 4);
  float*    logitbuf = (float*)carve((size_t)NC * 4);
  float*    hbuf     = (float*)carve((size_t)N * F * 4);  // reused as xagg later
  float*    tempx    = (float*)carve((size_t)N * F * 4);
  float*    supx     = (float*)carve((size_t)K * F * 4);
  float*    xagg     = hbuf;  // h is dead after tempx_final

  const int B = 256;

  // init accumulators (harness poisons ws once; we must zero every call)
  fill_f32_kernel<<<CDIV(N, B), B, 0, stream>>>(deg0, 0.0f, N);
  fill_f32_kernel<<<CDIV(N, B), B, 0, stream>>>(score, 0.0f, N);
  fill_f32_kernel<<<CDIV(N, B), B, 0, stream>>>(deg1, 0.0f, N);
  fill_f32_kernel<<<CDIV(N, B), B, 0, stream>>>(rowsum, 0.0f, N);
  fill_f32_kernel<<<CDIV(N, B), B, 0, stream>>>(zbuf, 0.0f, N);
  fill_u32_kernel<<<CDIV(N, B), B, 0, stream>>>((unsigned*)is_sup, 0u, N);
  fill_u32_kernel<<<CDIV(N, B), B, 0, stream>>>((unsigned*)node2sup, 0u, N);
  fill_ord_kernel<<<CDIV(N, B), B, 0, stream>>>(rowmax, -1e30f, N);
  fill_f32_kernel<<<CDIV(N * F, B), B, 0, stream>>>(tempx, 0.0f, N * F);

  // score GCN
  score_h1_kernel<<<CDIV(N, 8), B, 0, stream>>>(x, scoreW, h1);
  deg_w0_kernel<<<CDIV(E, B), B, 0, stream>>>(srcv, dstv, w, deg0);
  score_agg_kernel<<<CDIV(E, B), B, 0, stream>>>(srcv, dstv, w, deg0, h1, score);
  score_final_kernel<<<CDIV(SORT_N, B), B, 0, stream>>>(score, h1, deg0, scoreB, keys);

  // exact top-K: bitonic sort (descending) of packed keys
  for (int kk = 2; kk <= SORT_N; kk <<= 1)
    for (int j = kk >> 1; j > 0; j >>= 1)
      bitonic_step_kernel<<<SORT_N / B, B, 0, stream>>>(keys, j, kk);
  extract_topk_kernel<<<CDIV(K, B), B, 0, stream>>>(keys, sup_idx, is_sup, node2sup, batch, out);

  // h = x @ aggr_W  (WMMA f32 16x16x4, N/16 = 3125 tiles)
  gemm_h_kernel<<<N / 16, 256, 0, stream>>>(x, aggrW, hbuf);

  // aggregation GCN -> temp_x
  deg_w1_kernel<<<CDIV(E, B), B, 0, stream>>>(srcv, dstv, w, is_sup, w1buf, deg1);
  agg_edges_kernel<<<CDIV(E * 32, B), B, 0, stream>>>(srcv, dstv, w1buf, deg1, hbuf, tempx);
  tempx_final_kernel<<<CDIV(N * F, B), B, 0, stream>>>(tempx, hbuf, deg1, aggrB);

  // h dead now; reuse its buffer for x_agg
  fill_f32_kernel<<<CDIV(K * F, B), B, 0, stream>>>(xagg, 0.0f, K * F);
  supx_kernel<<<CDIV(K * F, B), B, 0, stream>>>(tempx, sup_idx, supx);

  // coarsened graph + assignment index outputs
  coarse_edges_kernel<<<CDIV(E, B), B, 0, stream>>>(srcv, dstv, w, is_sup, node2sup, out);
  assign_tail_kernel<<<CDIV(K, B), B, 0, stream>>>(sup_idx, out);

  // assignment weights: sigmoid(dot) -> row-norm -> gumbel softmax -> threshold
  aw_dot_kernel<<<CDIV(NC, 8), B, 0, stream>>>(tempx, supx, srcv, dstv, node2sup, sup_idx,
                                               w1buf, awbuf, rowsum);
  aw_norm_logit_kernel<<<CDIV(NC, B), B, 0, stream>>>(srcv, dstv, node2sup, sup_idx,
                                                      rowsum, temp, awbuf, logitbuf, rowmax);
  aw_exp_kernel<<<CDIV(NC, B), B, 0, stream>>>(srcv, dstv, node2sup, sup_idx,
                                               rowmax, awbuf, logitbuf, zbuf);
  aw_final_kernel<<<CDIV(NC, B), B, 0, stream>>>(srcv, dstv, node2sup, sup_idx,
                                                 zbuf, logitbuf, awbuf, out);

  // x_out = 0.5 * (sup_x + assign @ temp_x)
  xagg_kernel<<<CDIV(NC * 32, B), B, 0, stream>>>(srcv, dstv, node2sup, sup_idx,
                                                  awbuf, tempx, xagg);
  xout_kernel<<<CDIV(K * F, B), B, 0, stream>>>(supx, xagg, out);
}